// gat_k_73323681677635
// MI455X (gfx1250) — compile-verified
//
#include <hip/hip_runtime.h>

// GAT-style 3-hop graph attention, fp32, on CDNA5 WMMA (V_WMMA_F32_16X16X4_F32).
// The dominant 2048^3 batched GEMM (attn@attn) is LDS-tiled per workgroup with
// double-buffered async global->LDS staging (ASYNCcnt) on gfx1250.

typedef __attribute__((ext_vector_type(2))) float v2f;
typedef __attribute__((ext_vector_type(8))) float v8f;
typedef int v4i __attribute__((vector_size(16)));  // gcc-style, matches builtin proto

constexpr int B_ = 8, N_ = 2048, F_ = 128, U_ = 128;
constexpr float NEGV = -1.0e9f;

#define DEVFN __device__ __forceinline__

#if __has_builtin(__builtin_amdgcn_global_load_async_to_lds_b128)
#define ASYNC_LDS 1
#else
#define ASYNC_LDS 0
#endif

DEVFN v8f wmma4(v2f a, v2f b, v8f c) {
  // D = A(16x4) * B(4x16) + C(16x16), fp32
  return __builtin_amdgcn_wmma_f32_16x16x4_f32(
      false, a, false, b, (short)0, c, false, false);
}

// Fragment loader (ISA 7.12.2): lane-striped index (M for A, N for B) in lanes
// 0..15; K pairs (K0,K1) lanes 0-15 / (K2,K3) lanes 16-31 across the 2 VGPRs.
DEVFN v2f load_frag(const float* __restrict__ p, long sl, long sk, int i0, int k0) {
  const int lane = threadIdx.x & 31;
  const int half = lane >> 4;
  const int idx  = lane & 15;
  const float* q = p + (long)(i0 + idx) * sl + (long)(k0 + 2 * half) * sk;
  v2f v;
  v.x = q[0];
  v.y = q[sk];
  return v;
}

DEVFN v8f load_tile(const float* __restrict__ p, long ld, int m0, int n0) {
  const int lane = threadIdx.x & 31;
  const int half = lane >> 4;
  const int col  = lane & 15;
  v8f c;
#pragma unroll
  for (int r = 0; r < 8; ++r)
    c[r] = p[(long)(m0 + r + 8 * half) * ld + n0 + col];
  return c;
}

DEVFN void store_tile(float* __restrict__ p, long ld, int m0, int n0, v8f c, bool relu) {
  const int lane = threadIdx.x & 31;
  const int half = lane >> 4;
  const int col  = lane & 15;
#pragma unroll
  for (int r = 0; r < 8; ++r) {
    float v = c[r];
    if (relu) v = v > 0.0f ? v : 0.0f;
    p[(long)(m0 + r + 8 * half) * ld + n0 + col] = v;
  }
}

// 16B global -> LDS copy; async (ASYNCcnt) via gfx1250 GLOBAL_LOAD_ASYNC_TO_LDS_B128.
DEVFN void lds_cp16(const float* g, float* l) {
#if ASYNC_LDS
  typedef __attribute__((address_space(1))) v4i* gp_t;
  typedef __attribute__((address_space(3))) v4i* lp_t;
  __builtin_amdgcn_global_load_async_to_lds_b128(
      (gp_t)(unsigned long long)g,
      (lp_t)(unsigned int)(unsigned long long)l,
      0, 0);
#else
  *(float4*)l = *(const float4*)g;
#endif
}

DEVFN void async_wait0() {
#if ASYNC_LDS
  asm volatile("s_wait_asynccnt 0x0" ::: "memory");
#endif
}

// ---------------------------------------------------------------------------
// C[M,U] = A[M,F] @ W[F,U] (+ Cin), optional ReLU.  Wave tile 32x16 (2x1).
__global__ void k_gemm_fw(const float* __restrict__ A, const float* __restrict__ W,
                          const float* __restrict__ Cin, float* __restrict__ Cout,
                          int relu) {
  const int m0 = blockIdx.x * 32;
  const int n0 = blockIdx.y * 16;
  v8f acc0 = {}, acc1 = {};
  if (Cin) {
    acc0 = load_tile(Cin, U_, m0, n0);
    acc1 = load_tile(Cin, U_, m0 + 16, n0);
  }
#pragma unroll 4
  for (int k = 0; k < F_; k += 4) {
    v2f a0 = load_frag(A, F_, 1, m0, k);
    v2f a1 = load_frag(A, F_, 1, m0 + 16, k);
    v2f b  = load_frag(W, 1, U_, n0, k);
    acc0 = wmma4(a0, b, acc0);
    acc1 = wmma4(a1, b, acc1);
  }
  store_tile(Cout, U_, m0, n0, acc0, relu != 0);
  store_tile(Cout, U_, m0 + 16, n0, acc1, relu != 0);
}

// ---------------------------------------------------------------------------
// logits[b,n,m] = f[b,n,:]·ai[:,m] + f[b,m,:]·aj[:,n] + NEG*(1-|A[n,m]|)
__global__ void k_logits(const float* __restrict__ f, const float* __restrict__ ai,
                         const float* __restrict__ aj, const float* __restrict__ adj,
                         float* __restrict__ attn) {
  const int n0 = blockIdx.x * 16;
  const int m0 = blockIdx.y * 16;
  const int b  = blockIdx.z;
  const float* fb = f + (long)b * N_ * U_;
  v8f acc = {};
#pragma unroll 2
  for (int k = 0; k < U_; k += 4) {
    v2f aP = load_frag(fb, U_, 1, n0, k);
    v2f bP = load_frag(ai, 1, N_, m0, k);
    acc = wmma4(aP, bP, acc);
    v2f aQ = load_frag(aj, 1, N_, n0, k);
    v2f bQ = load_frag(fb, U_, 1, m0, k);
    acc = wmma4(aQ, bQ, acc);
  }
  const int lane = threadIdx.x & 31;
  const int half = lane >> 4;
  const int col  = lane & 15;
  float* ab = attn + (long)b * N_ * N_;
#pragma unroll
  for (int r = 0; r < 8; ++r) {
    const int n = n0 + r + 8 * half;
    const int m = m0 + col;
    const float Av = adj[(long)n * N_ + m] + (n == m ? 1.0f : 0.0f);
    ab[(long)n * N_ + m] = acc[r] + NEGV * (1.0f - fabsf(Av));
  }
}

// ---------------------------------------------------------------------------
// In-place column softmax over n (axis=1), then * (adj + I).
__global__ void k_softmax(float* __restrict__ attn, const float* __restrict__ adj) {
  const int m = blockIdx.x * blockDim.x + threadIdx.x;
  const int b = blockIdx.y;
  float* colp = attn + (long)b * N_ * N_ + m;
  float mx = -3.4e38f;
  for (int n = 0; n < N_; ++n) mx = fmaxf(mx, colp[(long)n * N_]);
  float s = 0.0f;
  for (int n = 0; n < N_; ++n) s += __expf(colp[(long)n * N_] - mx);
  const float inv = 1.0f / s;
  for (int n = 0; n < N_; ++n) {
    const float Av = adj[(long)n * N_ + m] + (n == m ? 1.0f : 0.0f);
    colp[(long)n * N_] = __expf(colp[(long)n * N_] - mx) * inv * Av;
  }
}

// ---------------------------------------------------------------------------
// h[b,m,j] = sum_n a[b,n,m] * x[b,n,j]   (aT @ x per batch)
// Wave tile 32x32 (2x2 register blocking): 4 WMMAs per 4 fragment loads.
__global__ void k_atx(const float* __restrict__ a, const float* __restrict__ x,
                      float* __restrict__ h) {
  const int m0 = blockIdx.x * 32;
  const int f0 = blockIdx.y * 32;
  const int b  = blockIdx.z;
  const float* ab = a + (long)b * N_ * N_;
  const float* xb = x + (long)b * N_ * F_;
  v8f c00 = {}, c01 = {}, c10 = {}, c11 = {};
#pragma unroll 2
  for (int k = 0; k < N_; k += 4) {
    v2f a0 = load_frag(ab, 1, N_, m0, k);       // A[i,k] = a[k, m0+i]
    v2f a1 = load_frag(ab, 1, N_, m0 + 16, k);
    v2f b0 = load_frag(xb, 1, F_, f0, k);       // B[k,j] = x[k, f0+j]
    v2f b1 = load_frag(xb, 1, F_, f0 + 16, k);
    c00 = wmma4(a0, b0, c00);
    c01 = wmma4(a0, b1, c01);
    c10 = wmma4(a1, b0, c10);
    c11 = wmma4(a1, b1, c11);
  }
  float* hb = h + (long)b * N_ * F_;
  store_tile(hb, F_, m0, f0, c00, false);
  store_tile(hb, F_, m0, f0 + 16, c01, false);
  store_tile(hb, F_, m0 + 16, f0, c10, false);
  store_tile(hb, F_, m0 + 16, f0 + 16, c11, false);
}

// ---------------------------------------------------------------------------
// a2 = attn @ attn per batch: 2048^3 GEMM.
// Workgroup = 4 waves, 64(M) x 128(N) tile; wave = 32x64 (2x4 WMMA blocking).
// K staged in LDS panels of 32, double-buffered, async global->LDS.
constexpr int KT  = 32;
constexpr int LDA = 36;   // padded A-panel row stride (floats): 16B aligned,
                          // 36*idx mod 64 distinct for idx 0..15 -> no conflicts
constexpr int LDB = 144;  // padded B-panel row stride: rows k/k+2 land on
                          // disjoint bank halves for the two lane groups

__global__ __launch_bounds__(128) void k_aa(const float* __restrict__ a,
                                            float* __restrict__ a2) {
  __shared__ float As[2][64 * LDA];
  __shared__ float Bs[2][KT * LDB];
  const int b  = blockIdx.z;
  const int n0 = blockIdx.x * 128;
  const int m0 = blockIdx.y * 64;
  const float* ab = a + (long)b * N_ * N_;
  const int tid  = threadIdx.x;
  const int wave = tid >> 5;
  const int wm = (wave >> 1) * 32;   // wave M offset inside the 64-row tile
  const int wn = (wave & 1) * 64;    // wave N offset inside the 128-col tile

  auto stage = [&](int kt, int buf) {
    const int k0 = kt * KT;
    // A panel: 64 rows x 32 K-floats = 512 float4 (8 per row)
#pragma unroll
    for (int i = 0; i < 4; ++i) {
      const int idx = tid + i * 128;
      const int ra = idx >> 3, ca = (idx & 7) * 4;
      lds_cp16(ab + (long)(m0 + ra) * N_ + k0 + ca, &As[buf][ra * LDA + ca]);
    }
    // B panel: 32 K-rows x 128 N-floats = 1024 float4 (32 per row)
#pragma unroll
    for (int i = 0; i < 8; ++i) {
      const int idx = tid + i * 128;
      const int rb = idx >> 5, cb = (idx & 31) * 4;
      lds_cp16(ab + (long)(k0 + rb) * N_ + n0 + cb, &Bs[buf][rb * LDB + cb]);
    }
  };

  v8f c[2][4];
#pragma unroll
  for (int i = 0; i < 2; ++i)
#pragma unroll
    for (int j = 0; j < 4; ++j) c[i][j] = {};

  stage(0, 0);
  async_wait0();
  __syncthreads();

  const int ntiles = N_ / KT;  // 64
  for (int kt = 0; kt < ntiles; ++kt) {
    const int cur = kt & 1;
    if (kt + 1 < ntiles) stage(kt + 1, cur ^ 1);  // overlap with compute
    const float* Ap = &As[cur][0];
    const float* Bp = &Bs[cur][0];
#pragma unroll
    for (int kk = 0; kk < KT; kk += 4) {
      v2f a0 = load_frag(Ap, LDA, 1, wm, kk);
      v2f a1 = load_frag(Ap, LDA, 1, wm + 16, kk);
      v2f bf[4];
#pragma unroll
      for (int j = 0; j < 4; ++j) bf[j] = load_frag(Bp, 1, LDB, wn + j * 16, kk);
#pragma unroll
      for (int j = 0; j < 4; ++j) {
        c[0][j] = wmma4(a0, bf[j], c[0][j]);
        c[1][j] = wmma4(a1, bf[j], c[1][j]);
      }
    }
    async_wait0();
    __syncthreads();
  }

  float* ob = a2 + (long)b * N_ * N_;
#pragma unroll
  for (int i = 0; i < 2; ++i)
#pragma unroll
    for (int j = 0; j < 4; ++j)
      store_tile(ob, N_, m0 + wm + i * 16, n0 + wn + j * 16, c[i][j], false);
}

// ---------------------------------------------------------------------------
extern "C" void kernel_launch(void* const* d_in, const int* /*in_sizes*/, int /*n_in*/,
                              void* d_out, int /*out_size*/, void* d_ws, size_t /*ws_size*/,
                              hipStream_t stream) {
  const float* x   = (const float*)d_in[0];   // [B,N,F]
  const float* adj = (const float*)d_in[1];   // [N,N]
  const float* w   = (const float*)d_in[2];   // [K,F,U]
  const float* ai  = (const float*)d_in[3];   // [U,N]
  const float* aj  = (const float*)d_in[4];   // [U,N]

  float* out_f    = (float*)d_out;                // relu(f): B*N*U
  float* out_attn = out_f + (long)B_ * N_ * U_;   // attn:    B*N*N

  float* f  = (float*)d_ws;                       // [B,N,U]  8 MB
  float* h  = f + (long)B_ * N_ * U_;             // [B,N,F]  8 MB
  float* a2 = h + (long)B_ * N_ * F_;             // [B,N,N]  128 MB

  const dim3 blk(32);
  const dim3 gFW(B_ * N_ / 32, U_ / 16);          // 512 x 8
  const dim3 gLG(N_ / 16, N_ / 16, B_);           // 128 x 128 x 8
  const dim3 gSM(N_ / 256, B_);                   // 8 x 8 (block 256)
  const dim3 gAX(N_ / 32, F_ / 32, B_);           // 64 x 4 x 8
  const dim3 gAA(N_ / 128, N_ / 64, B_);          // 16 x 32 x 8 (block 128)

  // hop 0: f = x @ w0
  k_gemm_fw<<<gFW, blk, 0, stream>>>(x, w, nullptr, f, 0);
  // attention logits + mask (in-place in d_out attn region)
  k_logits<<<gLG, blk, 0, stream>>>(f, ai, aj, adj, out_attn);
  // column softmax * (adj + I)
  k_softmax<<<gSM, dim3(256), 0, stream>>>(out_attn, adj);
  // hop 1: h = attn^T @ x ; f += h @ w1
  k_atx<<<gAX, blk, 0, stream>>>(out_attn, x, h);
  k_gemm_fw<<<gFW, blk, 0, stream>>>(h, w + (long)F_ * U_, f, f, 0);
  // hop 2: a2 = attn @ attn ; h = a2^T @ x ; out = relu(f + h @ w2)
  // (attn^3 in the reference is dead code and skipped)
  k_aa<<<gAA, dim3(128), 0, stream>>>(out_attn, a2);
  k_atx<<<gAX, blk, 0, stream>>>(a2, x, h);
  k_gemm_fw<<<gFW, blk, 0, stream>>>(h, w + 2L * F_ * U_, f, out_f, 1);
}